// EdgeUpdate_35871566856202
// MI455X (gfx1250) — compile-verified
//
#include <hip/hip_runtime.h>

typedef float v2f __attribute__((ext_vector_type(2)));
typedef float v4f __attribute__((ext_vector_type(4)));
typedef float v8f __attribute__((ext_vector_type(8)));

#define N_EDGES_C   1000000
#define N_DTILES    (N_EDGES_C/32)      // 31250 dual-tiles (32 edges each), exact
#define NWAVES      8
#define WG_THREADS  (NWAVES*32)
#define T_PER_WAVE  8                   // dual-tiles per wave -> 2048 edges per WG
#define DT_PER_WG   (NWAVES*T_PER_WAVE)

#define STRIDE      68                  // padded LDS row stride (bank-conflict free)
#define SBUF        (16*STRIDE)         // 1088 floats per stage buffer

// ---- LDS layout (floats) ----
#define L_WMLP   0                      // 192 tiles * 64 = 12288 (48KB)
#define L_WA     12288                  // 192 floats (+pad to 256)
#define L_STAGE  (L_WA + 256)           // per-wave: e0 + e1 + z = 3*SBUF
#define LDS_FLOATS (L_STAGE + NWAVES*3*SBUF)
#define LDS_BYTES  (LDS_FLOATS*4)       // 154624 B -> 2 WGs/WGP

#define WMMA4(A, B, C) __builtin_amdgcn_wmma_f32_16x16x4_f32(false, A, false, B, (short)0, C, false, false)

__device__ __forceinline__ float fast_tanh(float x) {
    float e = __expf(2.0f*x);
    return 1.0f - 2.0f/(e + 1.0f);
}
__device__ __forceinline__ float fast_sigmoid(float x) {
    return 1.0f / (1.0f + __expf(-x));
}

extern __shared__ float lds[];

// tiny pre-kernel: swizzle W_init into d_ws in B-fragment layout (lane-contiguous)
__global__ void swizzle_winit(const float* __restrict__ W_init, float* __restrict__ ws) {
    int tid = threadIdx.x;
    #pragma unroll
    for (int i = 0; i < 8; ++i) {
        int slot = tid + i*256;          // 64 tiles * 32 lanes
        int t = slot >> 5, L = slot & 31;
        int kk = ((t>>2)<<2) + ((L>>4)<<1);
        int n  = ((t&3)<<4) + (L&15);
        *(v2f*)(ws + t*64 + L*2) = *(const v2f*)(W_init + n*64 + kk);
    }
}

__global__ __launch_bounds__(WG_THREADS, 2)
void edge_update_kernel(const float* __restrict__ node_fts,
                        const float* __restrict__ edge_fts,
                        const int*   __restrict__ edges,     // [2, E]
                        const float* __restrict__ winit_swz, // d_ws, 4096 floats
                        const float* __restrict__ b_init,
                        const float* __restrict__ W_mlp,     // [64,192]
                        const float* __restrict__ b_mlp,
                        const float* __restrict__ W_alpha,   // [192]
                        const float* __restrict__ b_alpha,
                        float* __restrict__ out)
{
    const int tid  = threadIdx.x;
    const int lane = tid & 31;
    const int wave = tid >> 5;
    const int m    = lane & 15;
    const int h    = lane >> 4;

    // ---- cooperative swizzled W_mlp load into LDS (once per WG) ----
    #pragma unroll
    for (int i = 0; i < 24; ++i) {
        int slot = tid + i*WG_THREADS;   // 192 tiles * 32 lanes = 6144 slots
        int t = slot >> 5, L = slot & 31;
        int kk = ((t>>2)<<2) + ((L>>4)<<1);
        int n  = ((t&3)<<4) + (L&15);
        *(v2f*)(lds + L_WMLP + t*64 + L*2) = *(const v2f*)(W_mlp + n*192 + kk);
    }
    if (tid < 192) lds[L_WA + tid] = W_alpha[tid];
    __syncthreads();

    float* estage0 = lds + L_STAGE + wave*(3*SBUF);
    float* estage1 = estage0 + SBUF;
    float* zst     = estage0 + 2*SBUF;

    const float bi0 = b_init[m], bi1 = b_init[16+m], bi2 = b_init[32+m], bi3 = b_init[48+m];
    const float balpha0 = b_alpha[0];
    v4f bmlp[8];
    #pragma unroll
    for (int i = 0; i < 8; ++i) bmlp[i] = *(const v4f*)(b_mlp + h*32 + i*4);

    v8f acc0, acc1, acc2, acc3, acc4, acc5, acc6, acc7;

    auto dual_step = [&](v2f a0, v2f a1, const float* bb0, const float* bb1) {
        v2f b0 = *(const v2f*)(bb0);
        v2f b1 = *(const v2f*)(bb0 + 64);
        v2f b2 = *(const v2f*)(bb1);
        v2f b3 = *(const v2f*)(bb1 + 64);
        acc0 = WMMA4(a0, b0, acc0);  acc1 = WMMA4(a0, b1, acc1);
        acc2 = WMMA4(a0, b2, acc2);  acc3 = WMMA4(a0, b3, acc3);
        acc4 = WMMA4(a1, b0, acc4);  acc5 = WMMA4(a1, b1, acc5);
        acc6 = WMMA4(a1, b2, acc6);  acc7 = WMMA4(a1, b3, acc7);
    };

    auto epilogue = [&](const v8f& A0, const v8f& A1, const v8f& A2, const v8f& A3,
                        float alpha, const float* es, int row) {
        #pragma unroll
        for (int r = 0; r < 8; ++r) {            // stage z row-major (padded stride)
            int mm = (r + h*8)*STRIDE + m;
            zst[mm]      = A0[r];
            zst[mm + 16] = A1[r];
            zst[mm + 32] = A2[r];
            zst[mm + 48] = A3[r];
        }
        const float* zrow = zst + m*STRIDE + h*32;
        const float* erow = es  + m*STRIDE + h*32;
        float pv[32];
        float s1 = 0.f, s2 = 0.f;
        #pragma unroll
        for (int i = 0; i < 8; ++i) {
            v4f z4 = *(const v4f*)(zrow + i*4);
            #pragma unroll
            for (int j = 0; j < 4; ++j) {
                float p = fast_tanh(z4[j] + bmlp[i][j]);
                pv[i*4+j] = p;
                s1 += p; s2 += p*p;
            }
        }
        s1 += __shfl_xor(s1, 16);
        s2 += __shfl_xor(s2, 16);
        float mean = s1 * (1.0f/64.0f);
        float var  = s2 * (1.0f/64.0f) - mean*mean;
        float rsa  = rsqrtf(var + 1e-5f) * alpha;
        float t1 = 0.f, t2 = 0.f;
        #pragma unroll
        for (int i = 0; i < 8; ++i) {
            v4f e4 = *(const v4f*)(erow + i*4);
            #pragma unroll
            for (int j = 0; j < 4; ++j) {
                float s = e4[j] + (pv[i*4+j] - mean)*rsa;
                pv[i*4+j] = s;
                t1 += s; t2 += s*s;
            }
        }
        t1 += __shfl_xor(t1, 16);
        t2 += __shfl_xor(t2, 16);
        float mean2 = t1 * (1.0f/64.0f);
        float var2  = t2 * (1.0f/64.0f) - mean2*mean2;
        float rstd2 = rsqrtf(var2 + 1e-5f);
        float* orow = out + (size_t)row*64 + h*32;
        #pragma unroll
        for (int i = 0; i < 8; ++i) {
            v4f o;
            #pragma unroll
            for (int j = 0; j < 4; ++j) o[j] = (pv[i*4+j] - mean2)*rstd2;
            *(v4f*)(orow + i*4) = o;
        }
    };

    for (int t = 0; t < T_PER_WAVE; ++t) {
        const int dtile = blockIdx.x*DT_PER_WG + t*NWAVES + wave;
        if (dtile >= N_DTILES) break;
        const int e0   = dtile << 5;
        const int row0 = e0 + m;
        const int row1 = e0 + 16 + m;

        v8f vz = {};
        acc0=vz; acc1=vz; acc2=vz; acc3=vz; acc4=vz; acc5=vz; acc6=vz; acc7=vz;

        // ---- G1 (dual): e = edge_fts x W_init^T, B fragments from swizzled d_ws ----
        {
            const float* aptr0 = edge_fts + (size_t)row0*64 + h*2;
            const float* aptr1 = edge_fts + (size_t)row1*64 + h*2;
            const float* wib   = winit_swz + lane*2;
            #pragma unroll
            for (int ks = 0; ks < 16; ++ks) {
                v2f a0 = *(const v2f*)(aptr0 + ks*4);
                v2f a1 = *(const v2f*)(aptr1 + ks*4);
                dual_step(a0, a1, wib + ks*256, wib + ks*256 + 128);
            }
        }
        // bias + stage e (both sub-tiles), padded row stride
        #pragma unroll
        for (int r = 0; r < 8; ++r) {
            int mm = (r + h*8)*STRIDE + m;
            estage0[mm]      = acc0[r] + bi0;
            estage0[mm + 16] = acc1[r] + bi1;
            estage0[mm + 32] = acc2[r] + bi2;
            estage0[mm + 48] = acc3[r] + bi3;
            estage1[mm]      = acc4[r] + bi0;
            estage1[mm + 16] = acc5[r] + bi1;
            estage1[mm + 32] = acc6[r] + bi2;
            estage1[mm + 48] = acc7[r] + bi3;
        }

        // ---- G2 (dual): z = combined(32x192) x W_mlp^T ----
        acc0=vz; acc1=vz; acc2=vz; acc3=vz; acc4=vz; acc5=vz; acc6=vz; acc7=vz;
        float alf0 = 0.0f, alf1 = 0.0f;
        const float* h0 = node_fts + (size_t)((unsigned)edges[row0])*64 + h*2;
        const float* h1 = node_fts + (size_t)((unsigned)edges[row1])*64 + h*2;
        const float* t0 = node_fts + (size_t)((unsigned)edges[N_EDGES_C + row0])*64 + h*2;
        const float* t1 = node_fts + (size_t)((unsigned)edges[N_EDGES_C + row1])*64 + h*2;
        const float* eA0 = estage0 + m*STRIDE + h*2;
        const float* eA1 = estage1 + m*STRIDE + h*2;

        #pragma unroll
        for (int ks = 0; ks < 16; ++ks) {          // K 0..63 : e from LDS stage
            v2f a0 = *(const v2f*)(eA0 + ks*4);
            v2f a1 = *(const v2f*)(eA1 + ks*4);
            v2f wa = *(const v2f*)(lds + L_WA + ks*4 + h*2);
            alf0 += a0.x*wa.x + a0.y*wa.y;
            alf1 += a1.x*wa.x + a1.y*wa.y;
            const float* bb = lds + L_WMLP + ks*256 + lane*2;
            dual_step(a0, a1, bb, bb + 128);
        }
        #pragma unroll
        for (int ks = 0; ks < 16; ++ks) {          // K 64..127 : head gathers
            v2f a0 = *(const v2f*)(h0 + ks*4);
            v2f a1 = *(const v2f*)(h1 + ks*4);
            v2f wa = *(const v2f*)(lds + L_WA + 64 + ks*4 + h*2);
            alf0 += a0.x*wa.x + a0.y*wa.y;
            alf1 += a1.x*wa.x + a1.y*wa.y;
            const float* bb = lds + L_WMLP + (16+ks)*256 + lane*2;
            dual_step(a0, a1, bb, bb + 128);
        }
        #pragma unroll
        for (int ks = 0; ks < 16; ++ks) {          // K 128..191 : tail gathers
            v2f a0 = *(const v2f*)(t0 + ks*4);
            v2f a1 = *(const v2f*)(t1 + ks*4);
            v2f wa = *(const v2f*)(lds + L_WA + 128 + ks*4 + h*2);
            alf0 += a0.x*wa.x + a0.y*wa.y;
            alf1 += a1.x*wa.x + a1.y*wa.y;
            const float* bb = lds + L_WMLP + (32+ks)*256 + lane*2;
            dual_step(a0, a1, bb, bb + 128);
        }

        float alpha0 = fast_sigmoid(alf0 + __shfl_xor(alf0, 16) + balpha0);
        float alpha1 = fast_sigmoid(alf1 + __shfl_xor(alf1, 16) + balpha0);

        // sequential epilogues share the z-stage buffer (same-wave LDS is in-order)
        epilogue(acc0, acc1, acc2, acc3, alpha0, estage0, row0);
        epilogue(acc4, acc5, acc6, acc7, alpha1, estage1, row1);
    }
}

extern "C" void kernel_launch(void* const* d_in, const int* in_sizes, int n_in,
                              void* d_out, int out_size, void* d_ws, size_t ws_size,
                              hipStream_t stream) {
    (void)in_sizes; (void)n_in; (void)ws_size; (void)out_size;
    const float* node_fts = (const float*)d_in[0];
    const float* edge_fts = (const float*)d_in[1];
    const int*   edges    = (const int*)  d_in[2];
    const float* W_init   = (const float*)d_in[3];
    const float* b_init   = (const float*)d_in[4];
    const float* W_mlp    = (const float*)d_in[5];
    const float* b_mlp    = (const float*)d_in[6];
    const float* W_alpha  = (const float*)d_in[7];
    const float* b_alpha  = (const float*)d_in[8];
    float* out = (float*)d_out;
    float* wsw = (float*)d_ws;                 // 16KB: swizzled W_init

    swizzle_winit<<<dim3(1), dim3(256), 0, stream>>>(W_init, wsw);

    (void)hipFuncSetAttribute((const void*)edge_update_kernel,
                              hipFuncAttributeMaxDynamicSharedMemorySize, LDS_BYTES);
    dim3 grid((N_DTILES + DT_PER_WG - 1) / DT_PER_WG);   // 489
    dim3 block(WG_THREADS);
    edge_update_kernel<<<grid, block, LDS_BYTES, stream>>>(
        node_fts, edge_fts, edges, wsw, b_init, W_mlp, b_mlp, W_alpha, b_alpha, out);
}